// Model_34986803593317
// MI455X (gfx1250) — compile-verified
//
#include <hip/hip_runtime.h>

#define N_DIM 100
#define P_DIM 512
#define B_DIM 2048
#define NN    10000   // N*N

typedef float v2f __attribute__((ext_vector_type(2)));
typedef float v8f __attribute__((ext_vector_type(8)));

__device__ __forceinline__ float waveReduceSum(float v) {
    v += __shfl_xor(v, 16, 32);
    v += __shfl_xor(v,  8, 32);
    v += __shfl_xor(v,  4, 32);
    v += __shfl_xor(v,  2, 32);
    v += __shfl_xor(v,  1, 32);
    return v;
}

// ---------------------------------------------------------------------------
// K1: wg[r] = sum_j g[r,j] * w[j]^2 ; f[r] = exp(-wg[r])
// Memory-bound on g (400 MB). w^2 cached in LDS; one wave per row, float4 loads.
// ---------------------------------------------------------------------------
__global__ void wg_f_kernel(const float* __restrict__ g,
                            const float* __restrict__ w,
                            float* __restrict__ wg_out,
                            float* __restrict__ f_out) {
    __shared__ float wsh[NN];
    const int tid = threadIdx.x;
    for (int j = tid; j < NN; j += blockDim.x) {
        float wv = w[j];
        wsh[j] = wv * wv;
    }
    __syncthreads();

    const int wavesPerBlock = blockDim.x >> 5;
    const int wave  = blockIdx.x * wavesPerBlock + (tid >> 5);
    const int lane  = tid & 31;
    const int nWav  = gridDim.x * wavesPerBlock;

    for (int row = wave; row < NN; row += nWav) {
        const float4* g4 = (const float4*)(g + (size_t)row * NN);
        float acc = 0.f;
        for (int j = lane; j < NN / 4; j += 32) {
            float4 gv = g4[j];
            int p = 4 * j;
            acc += gv.x * wsh[p] + gv.y * wsh[p + 1] +
                   gv.z * wsh[p + 2] + gv.w * wsh[p + 3];
        }
        acc = waveReduceSum(acc);
        if (lane == 0) {
            wg_out[row] = acc;
            f_out[row]  = expf(-acc);
        }
    }
}

// ---------------------------------------------------------------------------
// K2: per b:  s[p] = x_mask[b,p,:]·alpha^2  (LDS),  c = 0.1*sum_p x_i*s,
//             V[b,n] = sum_p x[b,n,p]*s[p] - c
// Memory-bound on x (419 MB). One block (8 waves) per b; float4 loads of x.
// ---------------------------------------------------------------------------
__global__ void v_kernel(const float* __restrict__ x,
                         const float* __restrict__ x_mask,
                         const float* __restrict__ x_i,
                         float* __restrict__ V) {
    __shared__ float s_sh[P_DIM];
    __shared__ float red[256];
    __shared__ float c_sh;

    const int b   = blockIdx.x;
    const int tid = threadIdx.x;
    const float a0 = 0.001f * 0.001f;
    const float a1 = 0.002f * 0.002f;
    const float a2 = 0.004f * 0.004f;

    float cacc = 0.f;
    for (int p = tid; p < P_DIM; p += blockDim.x) {
        size_t mi = ((size_t)b * P_DIM + p) * 3;
        float s = x_mask[mi] * a0 + x_mask[mi + 1] * a1 + x_mask[mi + 2] * a2;
        s_sh[p] = s;
        cacc += x_i[(size_t)b * P_DIM + p] * s;
    }
    red[tid] = cacc;
    __syncthreads();
    for (int off = 128; off > 0; off >>= 1) {
        if (tid < off) red[tid] += red[tid + off];
        __syncthreads();
    }
    if (tid == 0) c_sh = 0.1f * red[0];
    __syncthreads();

    const int wave = tid >> 5;
    const int lane = tid & 31;
    for (int n = wave; n < N_DIM; n += 8) {
        const float4* xr = (const float4*)(x + ((size_t)b * N_DIM + n) * P_DIM);
        float acc = 0.f;
        for (int j = lane; j < P_DIM / 4; j += 32) {
            float4 xv = xr[j];
            int p = 4 * j;
            acc += xv.x * s_sh[p] + xv.y * s_sh[p + 1] +
                   xv.z * s_sh[p + 2] + xv.w * s_sh[p + 3];
        }
        acc = waveReduceSum(acc);
        if (lane == 0) V[(size_t)b * N_DIM + n] = acc - c_sh;
    }
}

// ---------------------------------------------------------------------------
// K3: Z[b,m] = sum_n f[m,n] * V[b,n] + 0.1*y_i[b,m]
// fp32 WMMA GEMM: per-wave 16(b) x 16(m) tile, K=100 in 25 steps of 4.
// A (16x4 f32): lane L holds V[bBase+(L&15), k0 + 2*(L>>4) + {0,1}]
// B (4x16 f32): lane L holds F[mBase+(L&15), k0 + 2*(L>>4) + {0,1}]
// 128 b-tiles * 7 m-tiles = 896 waves = 112 blocks x 8 waves. EXEC stays all-1s.
// ---------------------------------------------------------------------------
__global__ void z_wmma_kernel(const float* __restrict__ V,
                              const float* __restrict__ F,
                              const float* __restrict__ y_i,
                              float* __restrict__ Z) {
    const int wave = blockIdx.x * (blockDim.x >> 5) + (threadIdx.x >> 5);
    const int lane = threadIdx.x & 31;
    const int bTile = wave / 7;
    const int mTile = wave - bTile * 7;
    const int bBase = bTile * 16;
    const int mBase = mTile * 16;

    const int half = lane >> 4;   // 0: K=0,1   1: K=2,3
    const int lr   = lane & 15;

    const int  mCol   = mBase + lr;
    const bool mValid = (mCol < N_DIM);
    const int  mClamp = mValid ? mCol : (N_DIM - 1);
    const float bScale = mValid ? 1.f : 0.f;

    const float* aRow = V + (size_t)(bBase + lr) * N_DIM + 2 * half;
    const float* bRow = F + (size_t)mClamp * N_DIM + 2 * half;

    v8f c = {};
    #pragma unroll 5
    for (int k0 = 0; k0 < N_DIM; k0 += 4) {
        v2f a  = *(const v2f*)(aRow + k0);          // 8B aligned: rows*100 even, k even
        v2f bm = *(const v2f*)(bRow + k0);
        bm[0] *= bScale;
        bm[1] *= bScale;
        c = __builtin_amdgcn_wmma_f32_16x16x4_f32(
                /*neg_a=*/false, a, /*neg_b=*/false, bm,
                /*c_mod=*/(short)0, c, /*reuse_a=*/false, /*reuse_b=*/false);
    }

    if (mValid) {
        #pragma unroll
        for (int r = 0; r < 8; ++r) {
            int row = r + half * 8;
            size_t zi = (size_t)(bBase + row) * N_DIM + mCol;
            Z[zi] = c[r] + 0.1f * y_i[zi];
        }
    }
}

// ---------------------------------------------------------------------------
extern "C" void kernel_launch(void* const* d_in, const int* in_sizes, int n_in,
                              void* d_out, int out_size, void* d_ws, size_t ws_size,
                              hipStream_t stream) {
    (void)in_sizes; (void)n_in; (void)out_size; (void)ws_size;
    const float* x      = (const float*)d_in[0];   // [B, N, P]
    const float* x_mask = (const float*)d_in[1];   // [B, P, 3]
    const float* x_i    = (const float*)d_in[2];   // [B, P]
    const float* y_i    = (const float*)d_in[3];   // [B, N, 1]
    const float* g      = (const float*)d_in[4];   // [NN, NN]
    const float* w      = (const float*)d_in[5];   // [NN, 1]

    float* out = (float*)d_out;
    float* Z   = out;                              // [B, N]
    float* wg  = out + (size_t)B_DIM * N_DIM;      // [NN]

    float* ws = (float*)d_ws;
    float* F  = ws;                                // NN floats (exp(-wg), row-major NxN)
    float* V  = ws + 10016;                        // B*N floats, 16B-aligned offset

    wg_f_kernel<<<125, 256, 0, stream>>>(g, w, wg, F);
    v_kernel<<<B_DIM, 256, 0, stream>>>(x, x_mask, x_i, V);
    z_wmma_kernel<<<112, 256, 0, stream>>>(V, F, y_i, Z);
}